// OSDC_AMI_54434415510089
// MI455X (gfx1250) — compile-verified
//
#include <hip/hip_runtime.h>
#include <hip/hip_bf16.h>
#include <math.h>

#define BB   256
#define CC   3
#define TT   16384
#define LLW  20
#define RRW  20
#define NTH  1024

// ---- CDNA5 feature guards (device pass only) ----
#if defined(__HIP_DEVICE_COMPILE__) && defined(__gfx1250__) && \
    __has_builtin(__builtin_amdgcn_global_load_async_to_lds_b128)
#define HAVE_ASYNC 1
typedef __attribute__((__vector_size__(16))) int v4i;
typedef __attribute__((address_space(1))) v4i gv4i;   // global (__device__) AS
typedef __attribute__((address_space(3))) v4i lv4i;   // LDS (__shared__) AS
#else
#define HAVE_ASYNC 0
#endif

#if defined(__HIP_DEVICE_COMPILE__) && \
    __has_builtin(__builtin_amdgcn_wmma_f32_16x16x4_f32)
#define HAVE_WMMA4 1
typedef __attribute__((ext_vector_type(2))) float v2f;
typedef __attribute__((ext_vector_type(8))) float v8f;
#else
#define HAVE_WMMA4 0
#endif

// ---------- deterministic block reductions ----------
__device__ __forceinline__ float block_reduce_sum(float v, float* red, int tid) {
    red[tid] = v; __syncthreads();
    for (int s = NTH / 2; s > 0; s >>= 1) {
        if (tid < s) red[tid] += red[tid + s];
        __syncthreads();
    }
    float r = red[0]; __syncthreads();
    return r;
}

__device__ __forceinline__ float block_reduce_max(float v, float* red, int tid) {
    red[tid] = v; __syncthreads();
    for (int s = NTH / 2; s > 0; s >>= 1) {
        if (tid < s) red[tid] = fmaxf(red[tid], red[tid + s]);
        __syncthreads();
    }
    float r = red[0]; __syncthreads();
    return r;
}

// =====================================================================
// Kernel 1: windowed, class-weighted cross-entropy. One block per batch b.
// ws[b] = sum_t( w(t) * ce(t) * mask(t) ) / (B*T)
// =====================================================================
__global__ __launch_bounds__(NTH) void wce_kernel(
    const float* __restrict__ preds, const float* __restrict__ class_w,
    const float* __restrict__ mask, const int* __restrict__ label,
    float* __restrict__ ws) {
    __shared__ unsigned char lab[TT];   // 16 KB
    __shared__ float red[NTH];          // 4 KB
    const int b = blockIdx.x, tid = threadIdx.x;

    const int* lrow = label + (size_t)b * TT;
#pragma unroll
    for (int k = 0; k < TT / NTH; ++k) {
        const int t = tid + k * NTH;            // coalesced
        lab[t] = (unsigned char)lrow[t];
    }
    __syncthreads();

    const float cw0 = class_w[0], cw1 = class_w[1], cw2 = class_w[2];
    const float* pb   = preds + (size_t)b * CC * TT;
    const float* mrow = mask + (size_t)b * TT;

    float acc = 0.f;
    int   cnt = -1;                              // -1 => recompute window
    const int t0 = tid * 16;                     // 16 consecutive t per thread
    for (int j = 0; j < 16; ++j) {
        const int t = t0 + j;
        float w = 1.f;
        if (t >= LLW && t < TT - RRW) {
            if (cnt < 0) {
                cnt = 0;
                for (int q = t - LLW; q < t; ++q)
                    cnt += (int)((lab[q] != 0) != (lab[q + LLW] != 0));
            } else {
                cnt += (int)((lab[t - 1] != 0) != (lab[t - 1 + LLW] != 0))
                     - (int)((lab[t - 1 - LLW] != 0) != (lab[t - 1] != 0));
            }
            w = 0.1f * logf((float)cnt + 1.f) + 1.f;
        } else {
            cnt = -1;
        }
        const float p0 = pb[t], p1 = pb[TT + t], p2 = pb[2 * TT + t];
        const int lb = lab[t];
        const float m   = fmaxf(p0, fmaxf(p1, p2));
        const float lse = m + logf(expf(p0 - m) + expf(p1 - m) + expf(p2 - m));
        const float pl  = (lb == 0) ? p0 : ((lb == 1) ? p1 : p2);
        const float cw  = (lb == 0) ? cw0 : ((lb == 1) ? cw1 : cw2);
        const float ce  = -cw * (pl - lse);
        acc += w * ce * mrow[t];
    }
    const float tot = block_reduce_sum(acc, red, tid);
    if (tid == 0) ws[b] = tot * (1.f / ((float)BB * (float)TT));
}

// =====================================================================
// Kernel 2: temporal log-softmax MSE. One block per (b,c) row.
//   pm[t] = preds[row,t]*mask[b,t]
//   delta = LSE(pm[1:]) - LSE(pm[:-1])
//   ws[BB+row] = 0.25/(B*C*(T-1)) * sum_t min((pm[t+1]-pm[t]-delta)^2, 16)
// Row is staged in LDS via the CDNA5 async global->LDS path (ASYNCcnt).
// =====================================================================
__global__ __launch_bounds__(NTH) void mse_kernel(
    const float* __restrict__ preds, const float* __restrict__ mask,
    float* __restrict__ ws) {
    __shared__ float4 pmld4[TT / 4];    // 64 KB row (raw preds, then pm)
    __shared__ float  red[NTH];         // 4 KB
    float* pmld = (float*)pmld4;

    const int row = blockIdx.x;
    const int b   = row / CC;
    const int tid = threadIdx.x;
    const float* pr = preds + (size_t)row * TT;
    const float* mr = mask + (size_t)b * TT;

    float pmr[16];                      // this thread's 16 pm values (4 chunks)
#if HAVE_ASYNC
    // Issue async copies of the raw preds row into LDS: 16B per lane per call,
    // lanes contiguous -> fully coalesced 512B/wave requests. Tracked by ASYNCcnt.
#pragma unroll
    for (int k = 0; k < 4; ++k) {
        const int c = k * NTH + tid;
        __builtin_amdgcn_global_load_async_to_lds_b128(
            (gv4i*)(pr + 4 * c), (lv4i*)(&pmld4[c]), 0, 0);
    }
    // Overlap: fetch the mask row straight to registers while the copy runs.
    float4 mv[4];
#pragma unroll
    for (int k = 0; k < 4; ++k) mv[k] = ((const float4*)mr)[k * NTH + tid];
#if __has_builtin(__builtin_amdgcn_s_wait_asynccnt)
    __builtin_amdgcn_s_wait_asynccnt(0);
#else
    asm volatile("s_wait_asynccnt 0" ::: "memory");
#endif
    __syncthreads();
#pragma unroll
    for (int k = 0; k < 4; ++k) {
        const int c = k * NTH + tid;
        const float4 p = pmld4[c];
        float4 q;
        q.x = p.x * mv[k].x; q.y = p.y * mv[k].y;
        q.z = p.z * mv[k].z; q.w = p.w * mv[k].w;
        pmld4[c] = q;                    // masked row back to LDS for diff pass
        pmr[4 * k + 0] = q.x; pmr[4 * k + 1] = q.y;
        pmr[4 * k + 2] = q.z; pmr[4 * k + 3] = q.w;
    }
#else
#pragma unroll
    for (int k = 0; k < 4; ++k) {
        const int c = k * NTH + tid;
        const float4 p  = ((const float4*)pr)[c];
        const float4 m4 = ((const float4*)mr)[c];
        float4 q;
        q.x = p.x * m4.x; q.y = p.y * m4.y; q.z = p.z * m4.z; q.w = p.w * m4.w;
        pmld4[c] = q;
        pmr[4 * k + 0] = q.x; pmr[4 * k + 1] = q.y;
        pmr[4 * k + 2] = q.z; pmr[4 * k + 3] = q.w;
    }
#endif

    // Local maxima over [0,T-2] (for LSE0) and [1,T-1] (for LSE1).
    float mx0 = -3.4e38f, mx1 = -3.4e38f;
#pragma unroll
    for (int k = 0; k < 4; ++k)
#pragma unroll
        for (int e = 0; e < 4; ++e) {
            const int t = 4 * (k * NTH + tid) + e;
            const float v = pmr[4 * k + e];
            if (t < TT - 1) mx0 = fmaxf(mx0, v);
            if (t > 0)      mx1 = fmaxf(mx1, v);
        }
    const float M0 = block_reduce_max(mx0, red, tid);   // barrier also covers pmld stores
    const float M1 = block_reduce_max(mx1, red, tid);

    float s0 = 0.f, s1 = 0.f;
#pragma unroll
    for (int k = 0; k < 4; ++k)
#pragma unroll
        for (int e = 0; e < 4; ++e) {
            const int t = 4 * (k * NTH + tid) + e;
            const float v = pmr[4 * k + e];
            if (t < TT - 1) s0 += expf(v - M0);
            if (t > 0)      s1 += expf(v - M1);
        }
    const float S0 = block_reduce_sum(s0, red, tid);
    const float S1 = block_reduce_sum(s1, red, tid);
    const float delta = (M1 + logf(S1)) - (M0 + logf(S0));

    // Difference pass: strided t => LDS bank-conflict-free (bank = tid % 64).
    float ms = 0.f;
#pragma unroll
    for (int k = 0; k < TT / NTH; ++k) {
        const int t = tid + k * NTH;
        if (t < TT - 1) {
            float d = pmld[t + 1] - pmld[t] - delta;
            d = d * d;
            ms += fminf(d, 16.f);
        }
    }
    const float tot = block_reduce_sum(ms, red, tid);
    if (tid == 0)
        ws[BB + row] = tot * (0.25f / ((float)BB * (float)CC * (float)(TT - 1)));
}

// =====================================================================
// Kernel 3: deterministic final reduction of 1024 partials with a
// v_wmma_f32_16x16x4_f32 cross-lane sum (B = ones => D[m,n] = s_m + s_{m+16}).
// =====================================================================
__global__ __launch_bounds__(32) void finalize_kernel(
    const float* __restrict__ ws, float* __restrict__ out) {
    const int lane = threadIdx.x;       // exactly one wave32, EXEC all ones
    float s = 0.f;
    for (int k = 0; k < 32; ++k) s += ws[lane + 32 * k];  // fixed order
#if HAVE_WMMA4
    v2f a;  a[0] = s;   a[1] = 0.f;     // A[16x4]: col0 = s_{0..15}, col2 = s_{16..31}
    v2f bo; bo[0] = 1.f; bo[1] = 1.f;   // B = ones(4x16)
    v8f c = {};
    v8f d = __builtin_amdgcn_wmma_f32_16x16x4_f32(
        /*neg_a=*/false, a, /*neg_b=*/false, bo,
        /*c_mod=*/(short)0, c, /*reuse_a=*/false, /*reuse_b=*/false);
    float ls = 0.f;
#pragma unroll
    for (int i = 0; i < 8; ++i) ls += d[i];   // lane n<16: rows 0..7; n>=16: rows 8..15
    const float tot = __shfl(ls, 0, 32) + __shfl(ls, 16, 32);
#else
    float tot = s;
    for (int off = 16; off > 0; off >>= 1) tot += __shfl_down(tot, off, 32);
    tot = __shfl(tot, 0, 32);
#endif
    if (lane == 0) out[0] = tot;
}

// =====================================================================
extern "C" void kernel_launch(void* const* d_in, const int* in_sizes, int n_in,
                              void* d_out, int out_size, void* d_ws, size_t ws_size,
                              hipStream_t stream) {
    const float* preds   = (const float*)d_in[0];
    const float* class_w = (const float*)d_in[1];
    const float* mask    = (const float*)d_in[2];
    const int*   label   = (const int*)d_in[3];
    float* out = (float*)d_out;
    float* ws  = (float*)d_ws;          // [0..255]=wce partials, [256..1023]=mse partials

    wce_kernel<<<BB, NTH, 0, stream>>>(preds, class_w, mask, label, ws);
    mse_kernel<<<BB * CC, NTH, 0, stream>>>(preds, mask, ws);
    finalize_kernel<<<1, 32, 0, stream>>>(ws, out);
    (void)in_sizes; (void)n_in; (void)out_size; (void)ws_size;
}